// MoAT_80668075753500
// MI455X (gfx1250) — compile-verified
//
#include <hip/hip_runtime.h>
#include <hip/hip_bf16.h>

#define NN   256
#define LSTR 257          // padded LDS row stride (conflict-free columns)
#define EPSF 1e-7f

typedef float v2f __attribute__((ext_vector_type(2)));
typedef float v8f __attribute__((ext_vector_type(8)));

__device__ __forceinline__ float sigf(float x) { return 1.0f / (1.0f + __expf(-x)); }

// ---------------------------------------------------------------------------
// Blocked LU (no pivoting -- matrix is diagonally dominant) of a 256x256
// matrix living in LDS with row stride LSTR. Trailing (Schur) updates use
// V_WMMA_F32_16X16X4_F32, K=16 per block step as 4 chained WMMAs.
// The L panel is stored NEGATED in LDS (registers keep the true values for
// the solve recurrence) so the GEMM computes C + (-L)*U with no per-element
// negation; L entries are never re-read after their trailing update.
// Returns sum(log|U_ii|) valid on thread 0 only.
// Force-inlined so the compiler sees A in addrspace(3) and emits direct ds ops.
// ---------------------------------------------------------------------------
__device__ __forceinline__ float lu_logdet_lds(float* A)
{
    const int tid  = threadIdx.x;          // 0..255  (8 waves of 32)
    const int lane = tid & 31;
    const int wave = tid >> 5;
    const int half = (lane >> 4) & 1;      // 0: lanes 0-15, 1: lanes 16-31
    const int l15  = lane & 15;

    float ld = 0.0f;

    for (int k = 0; k < NN / 16; ++k) {
        const int kb = k * 16;

        // ---- 1) factor 16x16 diagonal block (serial, thread 0) ----
        if (tid == 0) {
            for (int p = 0; p < 16; ++p) {
                float piv = A[(kb + p) * LSTR + kb + p];
                ld += __logf(fabsf(piv));
                float inv = 1.0f / piv;
                for (int r = p + 1; r < 16; ++r) {
                    float m = A[(kb + r) * LSTR + kb + p] * inv;
                    A[(kb + r) * LSTR + kb + p] = m;
                    for (int c = p + 1; c < 16; ++c)
                        A[(kb + r) * LSTR + kb + c] -= m * A[(kb + p) * LSTR + kb + c];
                }
            }
        }
        __syncthreads();

        const int rem = NN - kb - 16;
        if (rem > 0) {
            // ---- 2) U panel: A[kb:kb+16, kb+16:] = Lkk^-1 * (.)  (thread/col) ----
            for (int j = kb + 16 + tid; j < NN; j += 256) {
                float xc[16];
                #pragma unroll
                for (int r = 0; r < 16; ++r) {
                    float v = A[(kb + r) * LSTR + j];
                    #pragma unroll
                    for (int p = 0; p < 16; ++p)
                        if (p < r) v -= A[(kb + r) * LSTR + kb + p] * xc[p];
                    xc[r] = v;
                    A[(kb + r) * LSTR + j] = v;
                }
            }
            // ---- 3) L panel: A[kb+16:, kb:kb+16] = (.) * Ukk^-1, stored negated ----
            for (int i = kb + 16 + tid; i < NN; i += 256) {
                float xr[16];
                #pragma unroll
                for (int c = 0; c < 16; ++c) {
                    float v = A[i * LSTR + kb + c];
                    #pragma unroll
                    for (int p = 0; p < 16; ++p)
                        if (p < c) v -= xr[p] * A[(kb + p) * LSTR + kb + c];
                    v /= A[(kb + c) * LSTR + kb + c];
                    xr[c] = v;                      // true value for recurrence
                    A[i * LSTR + kb + c] = -v;      // negated for fused C - L*U
                }
            }
            __syncthreads();

            // ---- 4) trailing update: C += (-L) * U, wave-owned 16x16 WMMA tiles ----
            // Loop bounds are wave-uniform => EXEC stays all-ones for WMMA.
            const int nt  = rem >> 4;
            const int ntt = nt * nt;
            for (int t = wave; t < ntt; t += 8) {
                const int Ib = kb + 16 + (t / nt) * 16;
                const int Jb = kb + 16 + (t % nt) * 16;

                // C tile, f32 16x16 layout: VGPR v -> (M = v + 8*half, N = l15)
                v8f c;
                #pragma unroll
                for (int v = 0; v < 8; ++v)
                    c[v] = A[(Ib + v + half * 8) * LSTR + Jb + l15];

                #pragma unroll
                for (int q = 0; q < 4; ++q) {
                    const int k0 = kb + q * 4 + half * 2;
                    // A frag 16x4: lane(M=l15) holds K = {k0, k0+1} (already -L)
                    v2f a, b;
                    a[0] = A[(Ib + l15) * LSTR + k0 + 0];
                    a[1] = A[(Ib + l15) * LSTR + k0 + 1];
                    // B frag 4x16: VGPR0 rows K={k0}, VGPR1 rows K={k0+1}, N = l15
                    b[0] = A[(k0 + 0) * LSTR + Jb + l15];
                    b[1] = A[(k0 + 1) * LSTR + Jb + l15];
                    c = __builtin_amdgcn_wmma_f32_16x16x4_f32(
                            false, a, false, b, (short)0, c, false, false);
                }

                #pragma unroll
                for (int v = 0; v < 8; ++v)
                    A[(Ib + v + half * 8) * LSTR + Jb + l15] = c[v];
            }
        }
        __syncthreads();
    }
    return ld;
}

// ---------------------------------------------------------------------------
// ws layout (floats): [0] = ld0 ; [16..271] = V0 ; [272..527] = V1 ;
//                     [528..528+65535] = Ws (symmetrized sigmoid(W))
// ---------------------------------------------------------------------------
__global__ __launch_bounds__(256) void prep_kernel(const float* __restrict__ W,
                                                   const float* __restrict__ Vc,
                                                   float* __restrict__ wsf)
{
    const int idx = blockIdx.x * 256 + threadIdx.x;   // 0..65535
    const int i = idx >> 8, j = idx & 255;
    float* V0 = wsf + 16;
    float* V1 = wsf + 272;
    float* Ws = wsf + 528;

    float w;
    if      (i > j) w = sigf(W[i * NN + j]);
    else if (i < j) w = sigf(W[j * NN + i]);   // symmetrize strict lower triangle
    else            w = 0.0f;
    Ws[i * NN + j] = w;

    if (idx < NN) {
        float s0 = sigf(Vc[idx * 2 + 0]);
        float s1 = sigf(Vc[idx * 2 + 1]);
        float inv = 1.0f / (s0 + s1);
        V0[idx] = s0 * inv;
        V1[idx] = s1 * inv;
    }
}

__global__ __launch_bounds__(256) void l0_kernel(float* __restrict__ wsf)
{
    __shared__ float A[NN * LSTR];
    const float* Ws = wsf + 528;
    const int i = threadIdx.x;

    if (i == 0) {                       // identity pad: det(A) = det(L0[1:,1:])
        A[0] = 1.0f;
        for (int j = 1; j < NN; ++j) A[j] = 0.0f;
    } else {
        float rs = 0.0f;
        for (int j = 0; j < NN; ++j) {
            float w = Ws[i * NN + j];
            rs += w;                            // diag = full row sum (incl. j=0)
            A[i * LSTR + j] = (j == 0 || j == i) ? 0.0f : -w;
        }
        A[i * LSTR + i] = rs;
    }
    __syncthreads();

    float ld = lu_logdet_lds(A);
    if (i == 0) wsf[0] = ld;
}

__global__ __launch_bounds__(256) void batch_kernel(const float* __restrict__ lambdas,
                                                    const int*   __restrict__ x,
                                                    const float* __restrict__ wsf,
                                                    float*       __restrict__ out)
{
    __shared__ float A[NN * LSTR];     // 256x257 fp32 = 257 KB of the 320 KB WGP LDS
    __shared__ float sPr[NN];
    __shared__ int   sx[NN];
    __shared__ float sred[8];

    const float* V0 = wsf + 16;
    const float* V1 = wsf + 272;
    const float* Ws = wsf + 528;

    const int b    = blockIdx.x;
    const int tid  = threadIdx.x;
    const int lane = tid & 31;
    const int wave = tid >> 5;

    // gather x row, per-node marginals Pr[i] = V[i, x_i]
    const int xi = x[b * NN + tid];
    sx[tid] = xi;
    const float pr = (xi == 0) ? V0[tid] : V1[tid];
    sPr[tid] = pr;

    // block reduction of sum(log Pr)
    float lp = __logf(pr);
    #pragma unroll
    for (int off = 16; off > 0; off >>= 1) lp += __shfl_down(lp, off);
    if (lane == 0) sred[wave] = lp;
    __syncthreads();

    // ---- build padded Laplacian A, one thread per row ----
    const int i = tid;
    if (i == 0) {
        A[0] = 1.0f;
        for (int j = 1; j < NN; ++j) A[j] = 0.0f;
    } else {
        const int   ai  = xi;
        const float pi0 = V0[i];
        const float pri = pr;
        float rs = 0.0f;
        for (int j = 0; j < NN; ++j) {
            float wp = 0.0f;
            if (j != i) {
                const float pj0   = V0[j];
                const float lower = fmaxf(EPSF, pi0 + pj0 - 1.0f);
                const float upper = fminf(pi0, pj0);
                const float s     = sigf(lambdas[i * NN + j]);
                const float P00   = lower + s * (upper - lower);
                float e;
                if (ai == 0) e = (sx[j] == 0) ? P00 : (pi0 - P00);
                else         e = (sx[j] == 0) ? (pj0 - P00) : (1.0f - pi0 - pj0 + P00);
                e = fminf(fmaxf(e, 0.0f), 1.0f);
                wp = Ws[i * NN + j] * (e / (pri * sPr[j]));
            }
            rs += wp;                                // diag includes j=0 term
            A[i * LSTR + j] = (j == 0 || j == i) ? 0.0f : -wp;   // col 0 padded
        }
        A[i * LSTR + i] = rs;
    }
    __syncthreads();

    const float ld = lu_logdet_lds(A);

    if (tid == 0) {
        float slp = 0.0f;
        #pragma unroll
        for (int w = 0; w < 8; ++w) slp += sred[w];
        out[b] = slp + ld - wsf[0];                  // + ld - ld0
    }
}

extern "C" void kernel_launch(void* const* d_in, const int* in_sizes, int n_in,
                              void* d_out, int out_size, void* d_ws, size_t ws_size,
                              hipStream_t stream)
{
    (void)in_sizes; (void)n_in; (void)out_size; (void)ws_size;
    const float* W       = (const float*)d_in[0];
    const float* lambdas = (const float*)d_in[1];
    const float* Vc      = (const float*)d_in[2];
    const int*   x       = (const int*)d_in[3];
    float* out = (float*)d_out;
    float* wsf = (float*)d_ws;

    prep_kernel <<<256,  256, 0, stream>>>(W, Vc, wsf);
    l0_kernel   <<<1,    256, 0, stream>>>(wsf);
    batch_kernel<<<1024, 256, 0, stream>>>(lambdas, x, wsf, out);
}